// SAGEConv2_30846455120745
// MI455X (gfx1250) — compile-verified
//
#include <hip/hip_runtime.h>

// ---------------------------------------------------------------------------
// GATConv-like reference collapses (rank-1 norm, heads==1) to:
//   deg[i]  = sum_e [row==i]                      (atomics over E)
//   dinv    = deg^-0.5
//   adv[i]  = sum_e [row==i] dinv[col[e]]         (atomics over E)
//   t[k]    = sum_i dinv[i] * x[i,k]              (GEMV, streams x 16MB)
//   v[k]    = sum_o att1[o] * lin_w[o,k]          (GEMV, streams lin_w 16MB)
//   s[j]    = sum_k t[k] * weight[k,j]            (GEMV, streams weight 16MB)
//   c2 = s.att2 ; cb = lin_b.att1
//   alpha[i]= lrelu(x[i,:].v + cb + adv[i]*c2)    (GEMV, streams x 16MB)
//   d  = s.alpha
//   out[i]  = relu(adv[i]*d + bias[i])
// Pure bandwidth problem (~72MB -> ~3.1us @ 23.3TB/s). GEMV inner loops run
// on v_wmma_f32_16x16x4_f32; K-loop unrolled 4x so 12 loads are in flight
// before the 4-WMMA accumulation group (staggered s_wait_loadcnt).
// ---------------------------------------------------------------------------

typedef float v2f __attribute__((ext_vector_type(2)));
typedef float v8f __attribute__((ext_vector_type(8)));

static __device__ __forceinline__ v8f wmma4(v2f a, v2f b, v8f c) {
  // (neg_a, A, neg_b, B, c_mod, C, reuse_a, reuse_b)
  return __builtin_amdgcn_wmma_f32_16x16x4_f32(false, a, false, b, (short)0, c,
                                               false, false);
}

// ---------------------------------------------------------------------------
__global__ void zero_kernel(float* p, int n) {
  int i = blockIdx.x * blockDim.x + threadIdx.x;
  if (i < n) p[i] = 0.0f;
}

__global__ void deg_kernel(const long long* __restrict__ ei, float* __restrict__ deg,
                           int E) {
  int e = blockIdx.x * blockDim.x + threadIdx.x;
  if (e < E) atomicAdd(&deg[(int)ei[e]], 1.0f);
}

__global__ void dinv_kernel(const float* __restrict__ deg, float* __restrict__ dinv,
                            int n) {
  int i = blockIdx.x * blockDim.x + threadIdx.x;
  if (i < n) dinv[i] = 1.0f / sqrtf(deg[i]);   // deg**-0.5 (deg==0 -> inf, matches)
}

__global__ void adv_kernel(const long long* __restrict__ ei,
                           const float* __restrict__ dinv, float* __restrict__ adv,
                           int E) {
  int e = blockIdx.x * blockDim.x + threadIdx.x;
  if (e < E) atomicAdd(&adv[(int)ei[e]], dinv[(int)ei[E + e]]);
}

// ---------------------------------------------------------------------------
// Column-space GEMV: out[j] (+)= sum_k vec[k] * M[k*C + j]
// One wave per (16-column tile, K-chunk). A carries vec replicated over the 16
// M-rows (lanes 0-15: K=k0,k0+1; lanes 16-31: K=k0+2,k0+3). B is the real
// 4x16 tile of M (VGPR0 = rows k0 / k0+2, VGPR1 = rows k0+1 / k0+3).
// D rows are all equal; lanes 0-15 emit column j0+lane.
// kchunk must be a multiple of 16 (all call sites: 1024 or 512).
__global__ void gemv_colspace_wmma(const float* __restrict__ vec,
                                   const float* __restrict__ M,
                                   float* __restrict__ out,
                                   int K, int C, int kchunk, int use_atomic) {
  const int lane = threadIdx.x;     // block = 1 wave of 32
  const int lc = lane & 15;
  const int h  = lane >> 4;         // half select (adds +2 to K index)
  const int j0 = blockIdx.x << 4;
  const int kb = blockIdx.y * kchunk;
  int ke = kb + kchunk; if (ke > K) ke = K;

  v8f c = {};
  for (int k0 = kb; k0 < ke; k0 += 16) {     // uniform loop -> EXEC all ones
    v2f a[4], b[4];
#pragma unroll
    for (int u = 0; u < 4; ++u) {            // issue all 12 loads up front
      const int k = k0 + 4 * u + 2 * h;
      a[u].x = vec[k];
      a[u].y = vec[k + 1];
      const float* mp = M + (size_t)k * C + j0 + lc;
      b[u].x = mp[0];
      b[u].y = mp[C];
    }
#pragma unroll
    for (int u = 0; u < 4; ++u) c = wmma4(a[u], b[u], c);
  }
  if (h == 0) {
    float r = c[0];                  // M=0 row, column j0+lc
    if (use_atomic) atomicAdd(&out[j0 + lc], r);
    else            out[j0 + lc] = r;
  }
}

// ---------------------------------------------------------------------------
// Row-space GEMV + epilogue: alpha[i] = lrelu( x[i,:].v + cb + adv[i]*c2 )
// A = 16 real rows of x (lane lc owns row i0+lc; float2 at K=k0+2h).
// B = v packed into column 0 only (lanes lc==0 carry it, others zero).
// Results: lane 0 holds rows i0..i0+7 in c[0..7], lane 16 holds i0+8..i0+15.
// K must be a multiple of 16 (K = IN = 512).
__global__ void alpha_rowspace_wmma(const float* __restrict__ x,
                                    const float* __restrict__ v,
                                    const float* __restrict__ adv,
                                    const float* __restrict__ scal, // [0]=c2 [1]=cb
                                    float* __restrict__ alpha, int K) {
  const int lane = threadIdx.x;
  const int lc = lane & 15;
  const int h  = lane >> 4;
  const int i0 = blockIdx.x << 4;
  const float* xr = x + (size_t)(i0 + lc) * K + 2 * h;
  const float msk = (lc == 0) ? 1.0f : 0.0f;   // branchless -> EXEC stays full

  v8f c = {};
  for (int k0 = 0; k0 < K; k0 += 16) {
    v2f a[4], b[4];
#pragma unroll
    for (int u = 0; u < 4; ++u) {
      const int k = k0 + 4 * u;
      a[u].x = xr[k];
      a[u].y = xr[k + 1];
      b[u].x = v[k + 2 * h]     * msk;
      b[u].y = v[k + 2 * h + 1] * msk;
    }
#pragma unroll
    for (int u = 0; u < 4; ++u) c = wmma4(a[u], b[u], c);
  }
  if (lc == 0) {                      // divergence only after the WMMA loop
    const float c2 = scal[0], cb = scal[1];
#pragma unroll
    for (int r = 0; r < 8; ++r) {
      int i = i0 + 8 * h + r;
      float z = c[r] + cb + adv[i] * c2;
      alpha[i] = (z >= 0.0f) ? z : 0.2f * z;
    }
  }
}

// ---------------------------------------------------------------------------
__global__ void dot_kernel(const float* __restrict__ a, const float* __restrict__ b,
                           float* __restrict__ out, int n) {
  __shared__ float sm[256];
  float acc = 0.0f;
  for (int i = threadIdx.x; i < n; i += 256) acc += a[i] * b[i];
  sm[threadIdx.x] = acc;
  __syncthreads();
  for (int s = 128; s > 0; s >>= 1) {
    if ((int)threadIdx.x < s) sm[threadIdx.x] += sm[threadIdx.x + s];
    __syncthreads();
  }
  if (threadIdx.x == 0) *out = sm[0];
}

__global__ void out_kernel(const float* __restrict__ adv, const float* __restrict__ bias,
                           const float* __restrict__ scal, float* __restrict__ out,
                           int n) {
  int i = blockIdx.x * blockDim.x + threadIdx.x;
  if (i < n) {
    float z = adv[i] * scal[2] + bias[i];
    out[i] = (z > 0.0f) ? z : 0.0f;
  }
}

// ---------------------------------------------------------------------------
extern "C" void kernel_launch(void* const* d_in, const int* in_sizes, int n_in,
                              void* d_out, int out_size, void* d_ws, size_t ws_size,
                              hipStream_t stream) {
  const float*     x      = (const float*)d_in[0];
  const long long* ei     = (const long long*)d_in[1];   // int64 per reference
  const float*     weight = (const float*)d_in[2];       // [IN, OUT] row-major
  const float*     bias   = (const float*)d_in[3];
  const float*     att    = (const float*)d_in[4];       // [2*OUT]
  const float*     lin_w  = (const float*)d_in[5];       // [OUT, IN] row-major
  const float*     lin_b  = (const float*)d_in[6];

  const int OUTC = in_sizes[3];               // 8192
  const int INC  = in_sizes[5] / OUTC;        // 512
  const int N    = in_sizes[0] / INC;         // 8192 (== OUTC by construction)
  const int E    = in_sizes[1] / 2;           // 262144

  float* ws    = (float*)d_ws;
  float* deg   = ws;                  // [N]   (atomic-accumulated)
  float* adv   = ws + N;              // [N]   (atomic-accumulated)
  float* dinv  = ws + 2 * N;          // [N]
  float* t     = ws + 3 * N;          // [INC] (atomic-accumulated)
  float* v     = t + INC;             // [INC] (atomic-accumulated)
  float* s     = v + INC;             // [OUTC]
  float* alpha = s + OUTC;            // [N]
  float* scal  = alpha + N;           // [0]=c2 [1]=cb [2]=d

  // Zero all atomic accumulators every call (graph replays reuse d_ws).
  const int zn = 3 * N + 2 * INC;
  zero_kernel<<<(zn + 255) / 256, 256, 0, stream>>>(ws, zn);

  // Edge segment-sums.
  deg_kernel<<<(E + 255) / 256, 256, 0, stream>>>(ei, deg, E);
  dinv_kernel<<<(N + 255) / 256, 256, 0, stream>>>(deg, dinv, N);
  adv_kernel<<<(E + 255) / 256, 256, 0, stream>>>(ei, dinv, adv, E);

  // t = dinv^T x            (K=N rows of x, C=INC cols; split K over 8 chunks)
  gemv_colspace_wmma<<<dim3(INC / 16, 8), 32, 0, stream>>>(
      dinv, x, t, N, INC, N / 8, 1);
  // v = att1^T lin_w        (K=OUTC rows of lin_w, C=INC cols)
  gemv_colspace_wmma<<<dim3(INC / 16, 8), 32, 0, stream>>>(
      att, lin_w, v, OUTC, INC, OUTC / 8, 1);
  // s = t @ weight          (K=INC rows of weight, C=OUTC cols; direct store)
  gemv_colspace_wmma<<<dim3(OUTC / 16, 1), 32, 0, stream>>>(
      t, weight, s, INC, OUTC, INC, 0);

  // Scalars: c2 = s.att2, cb = lin_b.att1
  dot_kernel<<<1, 256, 0, stream>>>(s, att + OUTC, &scal[0], OUTC);
  dot_kernel<<<1, 256, 0, stream>>>(lin_b, att, &scal[1], OUTC);

  // alpha = lrelu(x.v + cb + adv*c2)
  alpha_rowspace_wmma<<<N / 16, 32, 0, stream>>>(x, v, adv, scal, alpha, INC);

  // d = s.alpha ; out = relu(adv*d + bias)
  dot_kernel<<<1, 256, 0, stream>>>(s, alpha, &scal[2], N);
  out_kernel<<<(N + 255) / 256, 256, 0, stream>>>(adv, bias, scal, (float*)d_out, N);
}